// LigerFusedLinearCrossEntropyLossWithWeights_4655744549014
// MI455X (gfx1250) — compile-verified
//
#include <hip/hip_runtime.h>
#include <math.h>

#define IGNORE_INDEX (-100)
#define BT   2048
#define HDIM 4096
#define VDIM 32000
#define APITCH 4100   // 4096 + 4 floats: rows land on distinct LDS banks; 16B-aligned row starts

typedef float v2f __attribute__((ext_vector_type(2)));
typedef float v8f __attribute__((ext_vector_type(8)));

__global__ __launch_bounds__(256) void flce_main(
    const float* __restrict__ W,     // (V, H) row-major
    const float* __restrict__ X,     // (BT, H)
    const int*   __restrict__ tgt,   // (BT,)
    const float* __restrict__ bias,  // (V,)
    const float* __restrict__ lw,    // (BT,)
    float* __restrict__ t_out)       // (BT,) partial per-row losses
{
  __shared__ float sA[16 * APITCH];  // 262400 B — CDNA5 WGP has 320KB LDS
  __shared__ float redM[8 * 16];
  __shared__ float redS[8 * 16];
  __shared__ float goldSh[16];
  __shared__ int   maskSh[16];

  const int tid   = threadIdx.x;
  const int wave  = tid >> 5;
  const int lane  = tid & 31;
  const int nlane = lane & 15;   // N (and B-row) sub-index
  const int khalf = lane >> 4;   // lanes 16-31 carry K+2 (A/B 16x4 f32 layout)
  const int row0  = blockIdx.x * 16;

  // ---- stage 16 rows of X into LDS (float4, padded pitch) ----
  const int HV4 = HDIM / 4;
  for (int i = tid; i < 16 * HV4; i += 256) {
    int r = i / HV4, c = i - r * HV4;
    const float4 v = ((const float4*)(X + (size_t)(row0 + r) * HDIM))[c];
    *((float4*)&sA[r * APITCH + 4 * c]) = v;
  }
  __syncthreads();

  float mrun[8], srun[8];
  #pragma unroll
  for (int r = 0; r < 8; ++r) { mrun[r] = -__builtin_inff(); srun[r] = 0.f; }

  const float* aBase = &sA[nlane * APITCH + 2 * khalf];

  // Each wave processes PAIRS of adjacent 16-col tiles: one A fragment feeds
  // two independent WMMA accumulator chains (ILP for the 2-waves/SIMD regime).
  const int NPAIRS = VDIM / 32;  // 1000 -> exactly 125 pairs per wave, uniform EXEC
  for (int p = wave; p < NPAIRS; p += 8) {
    const int colBase = p * 32;
    const float* wBase0 = W + (size_t)(colBase + nlane) * HDIM + 2 * khalf;
    const float* wBase1 = wBase0 + (size_t)16 * HDIM;
    if (p + 8 < NPAIRS) {  // gfx1250 global_prefetch_b8, one pair ahead
      __builtin_prefetch(W + (size_t)(colBase + 256 + nlane) * HDIM + 2 * khalf, 0, 1);
      __builtin_prefetch(W + (size_t)(colBase + 272 + nlane) * HDIM + 2 * khalf, 0, 1);
    }

    v8f acc0 = {0.f, 0.f, 0.f, 0.f, 0.f, 0.f, 0.f, 0.f};
    v8f acc1 = {0.f, 0.f, 0.f, 0.f, 0.f, 0.f, 0.f, 0.f};
    #pragma unroll 4
    for (int kk = 0; kk < HDIM / 4; ++kk) {
      v2f a  = *(const v2f*)(aBase  + 4 * kk);  // ds_load_b64 (bank-conflict-free)
      v2f b0 = *(const v2f*)(wBase0 + 4 * kk);  // global_load_b64
      v2f b1 = *(const v2f*)(wBase1 + 4 * kk);
      acc0 = __builtin_amdgcn_wmma_f32_16x16x4_f32(false, a, false, b0,
                                                   (short)0, acc0, false, false);
      acc1 = __builtin_amdgcn_wmma_f32_16x16x4_f32(false, a, false, b1,
                                                   (short)0, acc1, false, false);
    }

    // ---- online softmax stats, both tiles fused per butterfly ----
    const float bv0 = bias[colBase + nlane];
    const float bv1 = bias[colBase + 16 + nlane];
    #pragma unroll
    for (int r = 0; r < 8; ++r) {
      float v0 = acc0[r] + bv0;                    // row = khalf*8 + r
      float v1 = acc1[r] + bv1;
      float mt = fmaxf(v0, v1);
      mt = fmaxf(mt, __shfl_xor(mt, 1, 32));
      mt = fmaxf(mt, __shfl_xor(mt, 2, 32));
      mt = fmaxf(mt, __shfl_xor(mt, 4, 32));
      mt = fmaxf(mt, __shfl_xor(mt, 8, 32));       // row-max across the 16-lane half
      float st = __expf(v0 - mt) + __expf(v1 - mt);
      st += __shfl_xor(st, 1, 32);
      st += __shfl_xor(st, 2, 32);
      st += __shfl_xor(st, 4, 32);
      st += __shfl_xor(st, 8, 32);                 // row sum-exp (32 cols)
      float mn = fmaxf(mrun[r], mt);
      srun[r] = srun[r] * __expf(mrun[r] - mn) + st * __expf(mt - mn);
      mrun[r] = mn;
    }
  }

  // ---- publish per-wave running stats (lanes 0 and 16 hold rows 0-7 / 8-15) ----
  if (nlane == 0) {
    #pragma unroll
    for (int r = 0; r < 8; ++r) {
      redM[wave * 16 + khalf * 8 + r] = mrun[r];
      redS[wave * 16 + khalf * 8 + r] = srun[r];
    }
  }
  __syncthreads();

  // ---- gold logit: 16 half-wave groups, one row each, X read from LDS ----
  {
    int g = tid >> 4, l = tid & 15;
    int tg  = tgt[row0 + g];
    int msk = (tg != IGNORE_INDEX);
    int srow = msk ? tg : 0;
    const float* wr = W + (size_t)srow * HDIM;
    float p = 0.f;
    #pragma unroll 8
    for (int j = l; j < HDIM; j += 16) p += sA[g * APITCH + j] * wr[j];
    p += __shfl_xor(p, 1, 32);
    p += __shfl_xor(p, 2, 32);
    p += __shfl_xor(p, 4, 32);
    p += __shfl_xor(p, 8, 32);
    if (l == 0) { goldSh[g] = p + bias[srow]; maskSh[g] = msk; }
  }
  __syncthreads();

  // ---- merge 8 waves' (m,s), emit per-row masked weighted (lse - gold) ----
  if (tid < 16) {
    const int row = tid;
    float M = -__builtin_inff();
    #pragma unroll
    for (int w = 0; w < 8; ++w) M = fmaxf(M, redM[w * 16 + row]);
    float S = 0.f;
    #pragma unroll
    for (int w = 0; w < 8; ++w) S += redS[w * 16 + row] * __expf(redM[w * 16 + row] - M);
    float lse = M + __logf(S);
    float tv = maskSh[row] ? (lse - goldSh[row]) * lw[row0 + row] : 0.f;
    t_out[row0 + row] = tv;
  }
}

__global__ __launch_bounds__(256) void flce_reduce(
    const float* __restrict__ t_in,   // (BT,)
    const int*   __restrict__ tgt,    // (BT,)
    const float* __restrict__ rs_p,   // scalar
    const int*   __restrict__ gas_p,  // scalar
    float* __restrict__ out)          // scalar
{
  __shared__ float sSum[256];
  __shared__ int   sCnt[256];
  const int tid = threadIdx.x;
  float s = 0.f; int c = 0;
  for (int i = tid; i < BT; i += 256) {
    s += t_in[i];
    c += (tgt[i] != IGNORE_INDEX) ? 1 : 0;
  }
  sSum[tid] = s; sCnt[tid] = c;
  __syncthreads();
  for (int off = 128; off > 0; off >>= 1) {
    if (tid < off) { sSum[tid] += sSum[tid + off]; sCnt[tid] += sCnt[tid + off]; }
    __syncthreads();
  }
  if (tid == 0) {
    float rs    = rs_p[0];
    float gas   = (float)gas_p[0];
    float n     = (float)(sCnt[0] > 0 ? sCnt[0] : 1);
    float total = sSum[0];
    float loss  = (rs == 0.f) ? 0.f : (total / n) / rs;
    out[0] = loss / gas;
  }
}

extern "C" void kernel_launch(void* const* d_in, const int* in_sizes, int n_in,
                              void* d_out, int out_size, void* d_ws, size_t ws_size,
                              hipStream_t stream) {
  const float* W    = (const float*)d_in[0];  // lin_weight (V,H)
  const float* X    = (const float*)d_in[1];  // _input (BT,H)
  const int*   tg   = (const int*)  d_in[2];  // target (BT,)
  const float* bias = (const float*)d_in[3];  // bias (V,)
  const float* lw   = (const float*)d_in[4];  // loss_weights (BT,)
  const float* rs   = (const float*)d_in[5];  // reduce_sum_loss_weights (scalar)
  const int*   gas  = (const int*)  d_in[6];  // grad_accumulation_steps (scalar)
  float* t = (float*)d_ws;                    // BT partial per-row losses

  flce_main  <<<BT / 16, 256, 0, stream>>>(W, X, tg, bias, lw, t);
  flce_reduce<<<1,       256, 0, stream>>>(t, tg, rs, gas, (float*)d_out);
}